// VectorQuantizer_20504173871414
// MI455X (gfx1250) — compile-verified
//
#include <hip/hip_runtime.h>
#include <hip/hip_bf16.h>

typedef __attribute__((ext_vector_type(2))) float v2f;
typedef __attribute__((ext_vector_type(8))) float v8f;

#define EMBED_DIM 64
#define CODEBOOK  1024
#define CHUNK     128          // codes staged in LDS at a time
#define ES        68           // LDS row stride (floats): even -> 8B aligned, avoids bank conflicts
#define ROWS_PER_BLOCK 128     // 8 waves * 16 rows

__global__ __launch_bounds__(256) void
vq_wmma_kernel(const float* __restrict__ X,    // [N,64]
               const float* __restrict__ E,    // [64,1024] row-major (D,K)
               float* __restrict__ quant,      // [N,64]
               float* __restrict__ enc,        // [N,1024] one-hot
               int*   __restrict__ idxOut,     // [N]
               int N)
{
    __shared__ float Et[CHUNK * ES];   // transposed codebook chunk: Et[code][d]
    __shared__ float En[CHUNK];        // ||e||^2 per staged code

    const int tid  = threadIdx.x;
    const int lane = tid & 31;
    const int wave = tid >> 5;
    const int ln   = lane & 15;        // N-index within 16 (C/B layout)
    const int hi   = lane >> 4;        // half-wave select (K pairs / row group)

    const int rowBase = blockIdx.x * ROWS_PER_BLOCK + wave * 16;

    // ---- Load A tile: 16 rows x 64 dims, f32 WMMA 16x4 layout ----
    // lane (ln,hi) holds, per k-chunk kc: K = kc*4 + 2*hi + {0,1} for row M=ln.
    int rowA = rowBase + ln;
    if (rowA >= N) rowA = N - 1;
    float2 a[16];
#pragma unroll
    for (int kc = 0; kc < 16; ++kc) {
        a[kc] = *(const float2*)&X[(size_t)rowA * EMBED_DIM + kc * 4 + 2 * hi];
    }

    // Running per-lane argmin over the codes this lane sees (its C column).
    float minD[8];
    int   minI[8];
#pragma unroll
    for (int r = 0; r < 8; ++r) { minD[r] = 3.402823466e38f; minI[r] = 0; }

    for (int ch = 0; ch < CODEBOOK / CHUNK; ++ch) {
        const int c0 = ch * CHUNK;

        // ---- Stage chunk transposed into LDS (coalesced global reads) ----
#pragma unroll
        for (int j = 0; j < (CHUNK * EMBED_DIM) / 256; ++j) {
            int i = tid + j * 256;              // 0 .. 8191
            int d = i >> 7;                      // dim
            int k = i & (CHUNK - 1);             // local code
            Et[k * ES + d] = E[(size_t)d * CODEBOOK + c0 + k];
        }
        __syncthreads();

        // ---- ||e||^2 for staged codes ----
        if (tid < CHUNK) {
            float s = 0.f;
#pragma unroll 8
            for (int d = 0; d < EMBED_DIM; ++d) {
                float v = Et[tid * ES + d];
                s += v * v;
            }
            En[tid] = s;
        }
        __syncthreads();

        // ---- 16-code tiles: chained f32 WMMA over K=64 ----
        for (int tt = 0; tt < CHUNK / 16; ++tt) {
            const int cl = tt * 16 + ln;                   // local code (= N column)
            const int cg = c0 + cl;                        // global code index
            const float2* Bp = (const float2*)&Et[cl * ES + 2 * hi];

            v8f acc = {};
#pragma unroll
            for (int kc = 0; kc < 16; ++kc) {
                float2 b = Bp[kc * 2];                     // Et[cl][kc*4 + 2*hi + {0,1}]
                v2f av; av[0] = a[kc].x; av[1] = a[kc].y;
                v2f bv; bv[0] = b.x;     bv[1] = b.y;
                acc = __builtin_amdgcn_wmma_f32_16x16x4_f32(
                    false, av, false, bv, (short)0, acc, false, false);
            }

            // distance (up to per-row constant) = ||e||^2 - 2*sim
            const float en = En[cl];
#pragma unroll
            for (int r = 0; r < 8; ++r) {
                float d = __builtin_fmaf(-2.0f, acc[r], en);
                if (d < minD[r]) { minD[r] = d; minI[r] = cg; }
            }
        }
        __syncthreads();
    }

    // ---- Final cross-lane argmin (16 lanes per half hold candidates per row),
    //      then write index, quantized gather, and fused one-hot. ----
#pragma unroll
    for (int r = 0; r < 8; ++r) {
        float dmin = minD[r];
        int   imin = minI[r];
#pragma unroll
        for (int m = 1; m < 16; m <<= 1) {
            float od = __shfl_xor(dmin, m, 32);
            int   oi = __shfl_xor(imin, m, 32);
            if (od < dmin || (od == dmin && oi < imin)) { dmin = od; imin = oi; }
        }
        const int row = rowBase + r + 8 * hi;   // lanes 0-15 -> row r, 16-31 -> r+8
        if (row < N) {
            if (ln == 0) idxOut[row] = imin;

            // quantized[row][:] = embeddings[:, imin]  (codebook lives in L2)
#pragma unroll
            for (int q = 0; q < 4; ++q) {
                int d = ln + 16 * q;
                quant[(size_t)row * EMBED_DIM + d] = E[(size_t)d * CODEBOOK + imin];
            }

            // one-hot row, zero-fill fused with the '1' write (coalesced float4)
#pragma unroll 4
            for (int j = 0; j < 16; ++j) {
                int base = j * 64 + ln * 4;
                float4 v; v.x = 0.f; v.y = 0.f; v.z = 0.f; v.w = 0.f;
                int rel = imin - base;
                if (rel >= 0 && rel < 4) (&v.x)[rel] = 1.0f;
                *(float4*)&enc[(size_t)row * CODEBOOK + base] = v;
            }
        }
    }
}

extern "C" void kernel_launch(void* const* d_in, const int* in_sizes, int n_in,
                              void* d_out, int out_size, void* d_ws, size_t ws_size,
                              hipStream_t stream) {
    const float* X = (const float*)d_in[0];   // [B,T,64] f32
    const float* E = (const float*)d_in[1];   // [64,1024] f32
    const int N = in_sizes[0] / EMBED_DIM;    // 131072

    float* out   = (float*)d_out;
    float* quant = out;                                   // N*64
    float* enc   = out + (size_t)N * EMBED_DIM;           // N*1024
    int*   idx   = (int*)(out + (size_t)N * EMBED_DIM + (size_t)N * CODEBOOK);

    const int blocks = (N + ROWS_PER_BLOCK - 1) / ROWS_PER_BLOCK;
    vq_wmma_kernel<<<blocks, 256, 0, stream>>>(X, E, quant, enc, idx, N);
}